// CompGraphConv_927712936002
// MI455X (gfx1250) — compile-verified
//
#include <hip/hip_runtime.h>
#include <stdint.h>

#define NN 50000      // nodes
#define NE 625000     // edges
#define DD 128        // feature dim
#define RR 200        // relations
#define BN_EPS 1e-5f

typedef float v2f __attribute__((ext_vector_type(2)));
typedef float v8f __attribute__((ext_vector_type(8)));

#define ATOMIC_ADD_F32(p, v) \
    __hip_atomic_fetch_add((p), (v), __ATOMIC_RELAXED, __HIP_MEMORY_SCOPE_AGENT)

// ---------------------------------------------------------------------------
// Kernel 1: edge scatter. One wave32 per edge; lane l owns features 4l..4l+3.
// comp = x[src] - r_feats[etype]*norm  ->  atomic add into agg_O / agg_I[dst]
// (linear projections are deferred to the node GEMM: segment_sum commutes
// with the linear maps, cutting GEMM FLOPs 12x).
// ---------------------------------------------------------------------------
__global__ __launch_bounds__(256) void edge_scatter_kernel(
    const float* __restrict__ x, const float* __restrict__ r_feats,
    const int* __restrict__ src, const int* __restrict__ dst,
    const int* __restrict__ etype, const float* __restrict__ norm,
    const uint8_t* __restrict__ out_mask,
    float* __restrict__ agg_O, float* __restrict__ agg_I,
    float* __restrict__ cnt_O, float* __restrict__ cnt_I)
{
    const int e    = blockIdx.x * 8 + (threadIdx.x >> 5);
    const int lane = threadIdx.x & 31;
    if (e >= NE) return;

    const int   s      = src[e];
    const int   d      = dst[e];
    const int   t      = etype[e];
    const float nm     = norm[e];
    const bool  is_out = (out_mask[e] != 0);

    const float4 xv = ((const float4*)(x       + (size_t)s * DD))[lane];
    const float4 rv = ((const float4*)(r_feats + (size_t)t * DD))[lane];
    float4 c;
    c.x = xv.x - rv.x * nm;
    c.y = xv.y - rv.y * nm;
    c.z = xv.z - rv.z * nm;
    c.w = xv.w - rv.w * nm;

    float* dp = (is_out ? agg_O : agg_I) + (size_t)d * DD + lane * 4;
    ATOMIC_ADD_F32(dp + 0, c.x);
    ATOMIC_ADD_F32(dp + 1, c.y);
    ATOMIC_ADD_F32(dp + 2, c.z);
    ATOMIC_ADD_F32(dp + 3, c.w);
    if (lane == 0) ATOMIC_ADD_F32((is_out ? cnt_O : cnt_I) + d, 1.0f);
}

// ---------------------------------------------------------------------------
// Kernel 2: fused node GEMM via V_WMMA_F32_16X16X4_F32 (full fp32 precision).
// Block = 256 threads = 8 waves; block handles 16 node rows, wave w handles
// output columns [16w, 16w+16). Three K=128 GEMMs accumulate into one tile:
//   (x - loop_rel) @ W_S.T  +  agg_O @ W_O.T  +  agg_I @ W_I.T
// Epilogue adds b_S + cnt_O*b_O + cnt_I*b_I and scales by 1/3.
// A frag (16x4 f32): lane<16 -> row=lane, K pair (k,k+1); lane>=16 -> +2.
// B frag (4x16 f32): B[k][n] = W[n][k] -> lane streams 2 consecutive W floats.
// ---------------------------------------------------------------------------
__global__ __launch_bounds__(256) void node_gemm_kernel(
    const float* __restrict__ x,     const float* __restrict__ loop_rel,
    const float* __restrict__ agg_O, const float* __restrict__ agg_I,
    const float* __restrict__ cnt_O, const float* __restrict__ cnt_I,
    const float* __restrict__ W_S,   const float* __restrict__ b_S,
    const float* __restrict__ W_O,   const float* __restrict__ b_O,
    const float* __restrict__ W_I,   const float* __restrict__ b_I,
    float* __restrict__ pre)
{
    const int lane = threadIdx.x & 31;
    const int wave = threadIdx.x >> 5;
    const int m0   = blockIdx.x * 16;        // node row base
    const int n0   = wave * 16;              // output col base
    const int mrow = lane & 15;
    const int ncol = lane & 15;
    const int koff = (lane >> 4) << 1;       // 0 for lanes 0-15, 2 for 16-31

    v8f acc = {};

    // GEMM 1: (x - loop_rel) @ W_S.T
    {
        const float* arow = x   + (size_t)(m0 + mrow) * DD;
        const float* brow = W_S + (size_t)(n0 + ncol) * DD;
        #pragma unroll 4
        for (int k = 0; k < DD; k += 4) {
            const int kk = k + koff;
            v2f a = *(const v2f*)(arow + kk) - *(const v2f*)(loop_rel + kk);
            v2f b = *(const v2f*)(brow + kk);
            acc = __builtin_amdgcn_wmma_f32_16x16x4_f32(
                      false, a, false, b, (short)0, acc, false, false);
        }
    }
    // GEMM 2: agg_O @ W_O.T
    {
        const float* arow = agg_O + (size_t)(m0 + mrow) * DD;
        const float* brow = W_O   + (size_t)(n0 + ncol) * DD;
        #pragma unroll 4
        for (int k = 0; k < DD; k += 4) {
            const int kk = k + koff;
            v2f a = *(const v2f*)(arow + kk);
            v2f b = *(const v2f*)(brow + kk);
            acc = __builtin_amdgcn_wmma_f32_16x16x4_f32(
                      false, a, false, b, (short)0, acc, false, false);
        }
    }
    // GEMM 3: agg_I @ W_I.T
    {
        const float* arow = agg_I + (size_t)(m0 + mrow) * DD;
        const float* brow = W_I   + (size_t)(n0 + ncol) * DD;
        #pragma unroll 4
        for (int k = 0; k < DD; k += 4) {
            const int kk = k + koff;
            v2f a = *(const v2f*)(arow + kk);
            v2f b = *(const v2f*)(brow + kk);
            acc = __builtin_amdgcn_wmma_f32_16x16x4_f32(
                      false, a, false, b, (short)0, acc, false, false);
        }
    }

    // Epilogue. C/D layout: VGPR r, lane l -> row = r + (l>=16 ? 8 : 0),
    // col = l & 15.
    const int   col   = n0 + ncol;
    const float bs    = b_S[col];
    const float bo    = b_O[col];
    const float bi    = b_I[col];
    const int   mbase = m0 + ((lane >> 4) << 3);
    #pragma unroll
    for (int r = 0; r < 8; ++r) {
        const int node = mbase + r;
        const float v = (acc[r] + bs + cnt_O[node] * bo + cnt_I[node] * bi)
                        * (1.0f / 3.0f);
        pre[(size_t)node * DD + col] = v;
    }
}

// ---------------------------------------------------------------------------
// Kernel 3: per-feature sum / sumsq reduction for BatchNorm batch stats.
// 128 threads per block (one per feature); each block strides over rows.
// ---------------------------------------------------------------------------
__global__ __launch_bounds__(128) void bn_reduce_kernel(
    const float* __restrict__ pre, float* __restrict__ sums)
{
    const int d = threadIdx.x;
    float s = 0.0f, sq = 0.0f;
    for (int row = blockIdx.x; row < NN; row += gridDim.x) {
        const float v = pre[(size_t)row * DD + d];
        s  += v;
        sq += v * v;
    }
    ATOMIC_ADD_F32(&sums[d],      s);
    ATOMIC_ADD_F32(&sums[DD + d], sq);
}

// ---------------------------------------------------------------------------
// Kernel 4: fold mean/var/gamma/beta into per-feature scale & shift.
// ---------------------------------------------------------------------------
__global__ __launch_bounds__(128) void bn_stats_kernel(
    const float* __restrict__ sums, const float* __restrict__ gamma,
    const float* __restrict__ beta, float* __restrict__ stats)
{
    const int d = threadIdx.x;
    const float inv_n = 1.0f / (float)NN;
    const float mean  = sums[d] * inv_n;
    const float var   = fmaxf(sums[DD + d] * inv_n - mean * mean, 0.0f);
    const float scale = rsqrtf(var + BN_EPS) * gamma[d];
    stats[d]      = scale;
    stats[DD + d] = beta[d] - mean * scale;
}

// ---------------------------------------------------------------------------
// Kernel 5: in-place normalize + tanh over the node output region.
// ---------------------------------------------------------------------------
__global__ __launch_bounds__(256) void bn_apply_kernel(
    float* __restrict__ out, const float* __restrict__ stats)
{
    const size_t i = (size_t)blockIdx.x * 256 + threadIdx.x;
    const int d = (int)(i & (DD - 1));
    out[i] = tanhf(out[i] * stats[d] + stats[DD + d]);
}

// ---------------------------------------------------------------------------
// Kernel 6: r_out = r_feats @ W_R.T + b_R  (tiny: 200x128x128)
// ---------------------------------------------------------------------------
__global__ __launch_bounds__(256) void rel_out_kernel(
    const float* __restrict__ r_feats, const float* __restrict__ W_R,
    const float* __restrict__ b_R, float* __restrict__ out)
{
    const int i = blockIdx.x * 256 + threadIdx.x;
    if (i >= RR * DD) return;
    const int r = i >> 7;
    const int n = i & (DD - 1);
    const float4* a = (const float4*)(r_feats + (size_t)r * DD);
    const float4* w = (const float4*)(W_R     + (size_t)n * DD);
    float s = b_R[n];
    #pragma unroll 8
    for (int k = 0; k < DD / 4; ++k) {
        const float4 av = a[k];
        const float4 wv = w[k];
        s += av.x * wv.x + av.y * wv.y + av.z * wv.z + av.w * wv.w;
    }
    out[i] = s;
}

// ---------------------------------------------------------------------------
extern "C" void kernel_launch(void* const* d_in, const int* in_sizes, int n_in,
                              void* d_out, int out_size, void* d_ws, size_t ws_size,
                              hipStream_t stream)
{
    (void)in_sizes; (void)n_in; (void)out_size; (void)ws_size;

    const float*   x        = (const float*)d_in[0];
    const float*   r_feats  = (const float*)d_in[1];
    const int*     src      = (const int*)d_in[2];
    const int*     dst      = (const int*)d_in[3];
    const int*     etype    = (const int*)d_in[4];
    const float*   norm     = (const float*)d_in[5];
    // d_in[6] = in_edges_mask (complement of out mask; unused)
    const uint8_t* out_mask = (const uint8_t*)d_in[7];
    const float*   W_O_w    = (const float*)d_in[8];
    const float*   W_O_b    = (const float*)d_in[9];
    const float*   W_I_w    = (const float*)d_in[10];
    const float*   W_I_b    = (const float*)d_in[11];
    const float*   W_S_w    = (const float*)d_in[12];
    const float*   W_S_b    = (const float*)d_in[13];
    const float*   W_R_w    = (const float*)d_in[14];
    const float*   W_R_b    = (const float*)d_in[15];
    const float*   loop_rel = (const float*)d_in[16];
    const float*   bn_gamma = (const float*)d_in[17];
    const float*   bn_beta  = (const float*)d_in[18];

    // workspace layout
    const size_t aggBytes = (size_t)NN * DD * sizeof(float);   // 25.6 MB each
    char*  ws    = (char*)d_ws;
    float* agg_O = (float*)(ws);
    float* agg_I = (float*)(ws + aggBytes);
    float* cnt_O = (float*)(ws + 2 * aggBytes);
    float* cnt_I = (float*)(ws + 2 * aggBytes + (size_t)NN * sizeof(float));
    float* sums  = (float*)(ws + 2 * aggBytes + 2 * (size_t)NN * sizeof(float));
    float* stats = sums + 2 * DD;
    const size_t zeroBytes = 2 * aggBytes + 2 * (size_t)NN * sizeof(float)
                           + 2 * DD * sizeof(float);

    float* outF  = (float*)d_out;
    float* pre   = outF;                       // node region, transformed in place
    float* r_out = outF + (size_t)NN * DD;

    hipMemsetAsync(d_ws, 0, zeroBytes, stream);

    edge_scatter_kernel<<<(NE + 7) / 8, 256, 0, stream>>>(
        x, r_feats, src, dst, etype, norm, out_mask,
        agg_O, agg_I, cnt_O, cnt_I);

    node_gemm_kernel<<<NN / 16, 256, 0, stream>>>(
        x, loop_rel, agg_O, agg_I, cnt_O, cnt_I,
        W_S_w, W_S_b, W_O_w, W_O_b, W_I_w, W_I_b, pre);

    bn_reduce_kernel<<<512, 128, 0, stream>>>(pre, sums);
    bn_stats_kernel<<<1, 128, 0, stream>>>(sums, bn_gamma, bn_beta, stats);
    bn_apply_kernel<<<((size_t)NN * DD) / 256, 256, 0, stream>>>(pre, stats);

    rel_out_kernel<<<(RR * DD + 255) / 256, 256, 0, stream>>>(
        r_feats, W_R_w, W_R_b, r_out);
}